// PoseGraphLoss_59631325938504
// MI455X (gfx1250) — compile-verified
//
#include <hip/hip_runtime.h>
#include <math.h>
#include <stdint.h>

#define EPSF 1e-8f
#define BS 256
#define NBLK 1024

typedef __attribute__((ext_vector_type(2))) float v2f;
typedef __attribute__((ext_vector_type(8))) float v8f;

// ---------- CDNA5 async global->LDS helpers ----------
template <int N>
__device__ __forceinline__ void wait_asynccnt() {
#if __has_builtin(__builtin_amdgcn_s_wait_asynccnt)
  __builtin_amdgcn_s_wait_asynccnt(N);
#else
  asm volatile("s_wait_asynccnt %0" ::"n"(N) : "memory");
#endif
}

__device__ __forceinline__ void async_b64(uint32_t lds_off, const void* gptr) {
  // GV mode: per-lane global addr (VGPR pair) -> per-lane LDS byte offset
  asm volatile("global_load_async_to_lds_b64 %0, %1, off"
               ::"v"(lds_off), "v"(gptr)
               : "memory");
}

__device__ __forceinline__ void stage_edge(uint32_t le,
                                           const int* __restrict__ edges,
                                           int e) {
  async_b64(le, (const char*)(edges + 2 * (size_t)e));  // 8B, 8-aligned
}

__device__ __forceinline__ void stage_meas(uint32_t lm,
                                           const float* __restrict__ meas,
                                           int e) {
  const char* gm = (const char*)(meas + 6 * (size_t)e);  // 24B, 8-aligned
  async_b64(lm + 0, gm + 0);
  async_b64(lm + 8, gm + 8);
  async_b64(lm + 16, gm + 16);
}

// ---------- fast native helpers (args are bounded -> HW trans units OK) ----
__device__ __forceinline__ float frcp(float x) { return __builtin_amdgcn_rcpf(x); }
__device__ __forceinline__ float fsqrt(float x) { return __builtin_amdgcn_sqrtf(x); }

// Branchless acos for x in [-1,1]: A&S 4.4.45, |err| <= 2e-8 (fp32-exact).
__device__ __forceinline__ float fast_acos(float x) {
  float ax = fabsf(x);
  float p = -0.0012624911f;
  p = fmaf(p, ax, 0.0066700901f);
  p = fmaf(p, ax, -0.0170881256f);
  p = fmaf(p, ax, 0.0308918810f);
  p = fmaf(p, ax, -0.0501743046f);
  p = fmaf(p, ax, 0.0889789874f);
  p = fmaf(p, ax, -0.2145988016f);
  p = fmaf(p, ax, 1.5707963050f);
  float r = fsqrt(1.0f - ax) * p;  // acos(|x|)
  return (x < 0.0f) ? (3.14159265358979f - r) : r;
}

// ---------- SE(3) math (all f32, W^2 = w w^T - |w|^2 I) ----------
__device__ __forceinline__ void exp_se3(const float xi[6], float R[9], float t[3]) {
  float wx = xi[0], wy = xi[1], wz = xi[2];
  float vx = xi[3], vy = xi[4], vz = xi[5];
  float t2 = wx * wx + wy * wy + wz * wz;
  float t2s = fmaxf(t2, 1e-12f);
  float th = fsqrt(t2s);
  float s = __sinf(th);  // native v_sin_f32 (|th| small, no range reduction)
  float c = __cosf(th);
  float rth = frcp(th);
  float rt2 = rth * rth;
  bool small = t2 < EPSF;
  float A = small ? 1.0f - t2 * (1.0f / 6.0f) : s * rth;
  float B = small ? 0.5f - t2 * (1.0f / 24.0f) : (1.0f - c) * rt2;
  float C = small ? (1.0f / 6.0f) - t2 * (1.0f / 120.0f) : (th - s) * rt2 * rth;
  float xx = wx * wx, yy = wy * wy, zz = wz * wz;
  float xy = wx * wy, xz = wx * wz, yz = wy * wz;
  // R = I + A*W + B*(w w^T - t2 I)
  R[0] = 1.0f + B * (xx - t2);
  R[1] = -A * wz + B * xy;
  R[2] = A * wy + B * xz;
  R[3] = A * wz + B * xy;
  R[4] = 1.0f + B * (yy - t2);
  R[5] = -A * wx + B * yz;
  R[6] = -A * wy + B * xz;
  R[7] = A * wx + B * yz;
  R[8] = 1.0f + B * (zz - t2);
  // V = I + B*W + C*(w w^T - t2 I)
  float V0 = 1.0f + C * (xx - t2), V1 = -B * wz + C * xy, V2 = B * wy + C * xz;
  float V3 = B * wz + C * xy, V4 = 1.0f + C * (yy - t2), V5 = -B * wx + C * yz;
  float V6 = -B * wy + C * xz, V7 = B * wx + C * yz, V8 = 1.0f + C * (zz - t2);
  t[0] = V0 * vx + V1 * vy + V2 * vz;
  t[1] = V3 * vx + V4 * vy + V5 * vz;
  t[2] = V6 * vx + V7 * vy + V8 * vz;
}

__device__ __forceinline__ float log_se3_sq(const float R[9], const float t[3]) {
  float tr = R[0] + R[4] + R[8];
  float cth = fminf(fmaxf(0.5f * (tr - 1.0f), -1.0f), 1.0f);
  float th = fast_acos(cth);
  float t2 = th * th;
  float t2s = fmaxf(t2, 1e-12f);
  bool small = t2 < EPSF;
  // sin(acos(c)) == sqrt(1-c^2), theta in [0,pi] so sin >= 0
  float sth = fsqrt(fmaxf(1.0f - cth * cth, 0.0f));
  float rth = frcp(th);
  float rt2s = frcp(t2s);
  float fac = small ? 0.5f + t2 * (1.0f / 12.0f) : 0.5f * th * frcp(sth);
  float w0 = fac * (R[7] - R[5]);
  float w1 = fac * (R[2] - R[6]);
  float w2 = fac * (R[3] - R[1]);
  float A = small ? 1.0f : sth * rth;
  float B = small ? 0.5f : (1.0f - cth) * rt2s;
  float coef = small ? (1.0f / 12.0f) + t2 * (1.0f / 720.0f)
                     : (1.0f - 0.5f * A * frcp(B)) * rt2s;
  float wdot = w0 * w0 + w1 * w1 + w2 * w2;
  // Vinv = I - 0.5*W + coef*(w w^T - wdot I)
  float diag = 1.0f - coef * wdot;
  float V00 = diag + coef * w0 * w0;
  float V01 = 0.5f * w2 + coef * w0 * w1;
  float V02 = -0.5f * w1 + coef * w0 * w2;
  float V10 = -0.5f * w2 + coef * w0 * w1;
  float V11 = diag + coef * w1 * w1;
  float V12 = 0.5f * w0 + coef * w1 * w2;
  float V20 = 0.5f * w1 + coef * w0 * w2;
  float V21 = -0.5f * w0 + coef * w1 * w2;
  float V22 = diag + coef * w2 * w2;
  float v0 = V00 * t[0] + V01 * t[1] + V02 * t[2];
  float v1 = V10 * t[0] + V11 * t[1] + V12 * t[2];
  float v2 = V20 * t[0] + V21 * t[1] + V22 * t[2];
  return wdot + v0 * v0 + v1 * v1 + v2 * v2;
}

__device__ __forceinline__ float edge_loss(const float* __restrict__ poses,
                                           int i, int j, const float m[6]) {
  // pose rows are 24B (8-aligned) -> 3x float2 loads each
  float xi[6], xj[6];
  const float2* pi2 = (const float2*)(poses + 6 * (size_t)i);
  const float2* pj2 = (const float2*)(poses + 6 * (size_t)j);
#pragma unroll
  for (int k = 0; k < 3; ++k) {
    float2 a = pi2[k];
    xi[2 * k] = a.x;
    xi[2 * k + 1] = a.y;
    float2 b = pj2[k];
    xj[2 * k] = b.x;
    xj[2 * k + 1] = b.y;
  }
  float Ri[9], ti[3], Rj[9], tj[3], Rm[9], tm[3];
  exp_se3(xi, Ri, ti);
  exp_se3(xj, Rj, tj);
  exp_se3(m, Rm, tm);
  float d0 = tj[0] - ti[0], d1 = tj[1] - ti[1], d2 = tj[2] - ti[2];
  // tc = Ri^T d
  float tc0 = Ri[0] * d0 + Ri[3] * d1 + Ri[6] * d2;
  float tc1 = Ri[1] * d0 + Ri[4] * d1 + Ri[7] * d2;
  float tc2 = Ri[2] * d0 + Ri[5] * d1 + Ri[8] * d2;
  // Rc = Ri^T Rj
  float Rc[9];
#pragma unroll
  for (int a = 0; a < 3; ++a)
#pragma unroll
    for (int c = 0; c < 3; ++c)
      Rc[a * 3 + c] = Ri[0 + a] * Rj[0 + c] + Ri[3 + a] * Rj[3 + c] +
                      Ri[6 + a] * Rj[6 + c];
  // Rr = Rm^T Rc
  float Rr[9];
#pragma unroll
  for (int a = 0; a < 3; ++a)
#pragma unroll
    for (int c = 0; c < 3; ++c)
      Rr[a * 3 + c] = Rm[0 + a] * Rc[0 + c] + Rm[3 + a] * Rc[3 + c] +
                      Rm[6 + a] * Rc[6 + c];
  // tr = Rm^T (tc - tm)
  float u0 = tc0 - tm[0], u1 = tc1 - tm[1], u2 = tc2 - tm[2];
  float trv[3];
  trv[0] = Rm[0] * u0 + Rm[3] * u1 + Rm[6] * u2;
  trv[1] = Rm[1] * u0 + Rm[4] * u1 + Rm[7] * u2;
  trv[2] = Rm[2] * u0 + Rm[5] * u1 + Rm[8] * u2;
  return log_se3_sq(Rr, trv);
}

// ---------- main kernel ----------
// Pipeline (per thread, tiles e_k = e0 + k*stride):
//   edges staged 2 tiles ahead (triple buffer), measurements 1 tile ahead
//   (double buffer) -> 4 async ops/iter, s_wait_asynccnt<=4 guarantees tile k
//   data + tile k+1 indices resident (async loads complete in order).
//   Tile k+1 indices feed global_prefetch of the next pose rows, hiding the
//   random-gather latency behind ~600 VALU/TRANS ops of SE(3) math.
__global__ __launch_bounds__(BS) void pg_loss_kernel(
    const float* __restrict__ poses, const int* __restrict__ edges,
    const float* __restrict__ meas, float* __restrict__ partials, int E) {
  __shared__ int2 s_edge[3][BS];
  __shared__ float s_meas[2][BS][6];
  __shared__ float s_red[BS / 32];

  const int tid = threadIdx.x;
  const int stride = gridDim.x * blockDim.x;
  const int e0 = blockIdx.x * blockDim.x + tid;

  // LDS byte offsets for this thread's slots (low 32 bits of generic ptr == AS3 offset)
  uint32_t le[3], lm[2];
#pragma unroll
  for (int b = 0; b < 3; ++b) le[b] = (uint32_t)(uintptr_t)&s_edge[b][tid];
#pragma unroll
  for (int b = 0; b < 2; ++b) lm[b] = (uint32_t)(uintptr_t)&s_meas[b][tid][0];

  float acc = 0.0f;
  if (e0 < E) {
    // prologue: E(0), E(1), M(0)  -> 5 async ops
    int e1 = e0 + stride;
    int e1c = (e1 < E) ? e1 : e0;
    stage_edge(le[0], edges, e0);
    stage_edge(le[1], edges, e1c);
    stage_meas(lm[0], meas, e0);

    int c0 = 0, c1 = 1, c2 = 2;  // rotating edge-buffer indices (cur/next/far)
    int mb = 0;                  // measurement ping-pong
    for (int e = e0; e < E; e += stride) {
      int en1 = e + stride;
      int en2 = en1 + stride;
      int en1c = (en1 < E) ? en1 : e;
      int en2c = (en2 < E) ? en2 : en1c;
      stage_edge(le[c2], edges, en2c);      // E(k+2)
      stage_meas(lm[mb ^ 1], meas, en1c);   // M(k+1) : 3 ops
      wait_asynccnt<4>();                   // E(k),E(k+1),M(k) resident

      int2 ij = s_edge[c0][tid];
      int2 ijn = s_edge[c1][tid];
      __builtin_prefetch(poses + 6 * (size_t)ijn.x, 0, 3);  // global_prefetch_b8
      __builtin_prefetch(poses + 6 * (size_t)ijn.y, 0, 3);
      float m[6];
#pragma unroll
      for (int k = 0; k < 6; ++k) m[k] = s_meas[mb][tid][k];

      acc += edge_loss(poses, ij.x, ij.y, m);

      int t = c0;  c0 = c1;  c1 = c2;  c2 = t;  // rotate edge buffers
      mb ^= 1;
    }
    wait_asynccnt<0>();  // drain before leaving the pipeline
  }

  // wave32 butterfly reduction (deterministic), then 8 per-wave partials
#pragma unroll
  for (int off = 16; off > 0; off >>= 1) acc += __shfl_down(acc, off, 32);
  if ((tid & 31) == 0) s_red[tid >> 5] = acc;
  __syncthreads();
  if (tid == 0) {
    float s = 0.0f;
#pragma unroll
    for (int w = 0; w < BS / 32; ++w) s += s_red[w];
    partials[blockIdx.x] = s;
  }
}

// ---------- finalize: WMMA ones-GEMM reduction of the block partials -------
// Single wave (EXEC all-ones as WMMA requires). Each 64-float chunk of
// `partials` is a 4x16 B block; D = A(16x4 ones) * B + C accumulates the 16
// running column sums (replicated across rows). 16 chained WMMAs cover
// NBLK=1024 partials; a 16-lane shuffle tree finishes. Deterministic:
// fixed-function, fixed order.
__global__ __launch_bounds__(32) void pg_finalize(
    const float* __restrict__ partials, float* __restrict__ out,
    double inv_e) {
  const int lane = threadIdx.x;
  v2f a;
  a.x = 1.0f;
  a.y = 1.0f;
  v8f c = {};
  // B layout (4x16 f32): lanes 0-15 hold K=0,1 for N=lane; lanes 16-31 K=2,3.
  const int col = (lane < 16) ? lane : (lane + 16);
#pragma unroll
  for (int base = 0; base < NBLK; base += 64) {
    v2f b;
    b.x = partials[base + col];
    b.y = partials[base + col + 16];
    c = __builtin_amdgcn_wmma_f32_16x16x4_f32(false, a, false, b, (short)0, c,
                                              false, false);
  }
  // c[0] on lane n (n<16) = total column sum of N=n; rows replicated.
  float s = c[0];
  s += __shfl_down(s, 8, 16);
  s += __shfl_down(s, 4, 16);
  s += __shfl_down(s, 2, 16);
  s += __shfl_down(s, 1, 16);
  if (lane == 0) out[0] = (float)((double)s * inv_e);
}

extern "C" void kernel_launch(void* const* d_in, const int* in_sizes, int n_in,
                              void* d_out, int out_size, void* d_ws,
                              size_t ws_size, hipStream_t stream) {
  (void)n_in;
  (void)out_size;
  (void)ws_size;
  const float* poses = (const float*)d_in[0];
  const int* edges = (const int*)d_in[1];
  const float* meas = (const float*)d_in[2];
  const int E = in_sizes[1] / 2;  // edges is (E, 2)
  float* partials = (float*)d_ws;

  pg_loss_kernel<<<NBLK, BS, 0, stream>>>(poses, edges, meas, partials, E);
  pg_finalize<<<1, 32, 0, stream>>>(partials, (float*)d_out, 1.0 / (double)E);
}